// DropoutGCN_11905649344776
// MI455X (gfx1250) — compile-verified
//
#include <hip/hip_runtime.h>

typedef float v2f __attribute__((ext_vector_type(2)));
typedef float v8f __attribute__((ext_vector_type(8)));

// ---------------------------------------------------------------------------
// Generic skinny GEMM via V_WMMA_F32_16X16X4_F32.
// C[M x Nc] = A[M x K] (row major) @ B[K x Nc] (row major) + bias, optional relu.
// One wave computes one 16x16 output tile.
// A-operand layout (16x4 f32): lanes 0-15 hold K=k+0/k+1, lanes 16-31 K=k+2/k+3.
// B-operand layout (4x16 f32): vgpr0 = row (k+2*half), vgpr1 = row (k+2*half+1),
//   column = lane&15.  C/D: vgpr r -> row m0+r (lanes 0-15) / m0+8+r (lanes 16-31).
// ---------------------------------------------------------------------------
__global__ void gk_gemm_wmma_f32(const float* __restrict__ A,
                                 const float* __restrict__ B,
                                 const float* __restrict__ bias,
                                 float* __restrict__ C,
                                 int M, int K, int Nc, int relu)
{
    int gtid = blockIdx.x * blockDim.x + threadIdx.x;
    int wave = gtid >> 5;
    int lane = threadIdx.x & 31;
    int colTiles = Nc >> 4;
    int rowTiles = (M + 15) >> 4;
    if (wave >= rowTiles * colTiles) return;          // wave-uniform: EXEC stays full
    int mt = wave / colTiles;
    int nt = wave - mt * colTiles;
    int m0 = mt << 4, n0 = nt << 4;
    int half = lane >> 4;
    int l    = lane & 15;

    int arow = m0 + l;
    if (arow >= M) arow = M - 1;                      // clamp; garbage rows not stored
    const float* Arow = A + (size_t)arow * K;

    v8f acc = {0.f, 0.f, 0.f, 0.f, 0.f, 0.f, 0.f, 0.f};
    for (int k = 0; k < K; k += 4) {
        v2f a, b;
        a.x = Arow[k + 2 * half];
        a.y = Arow[k + 2 * half + 1];
        const float* bp = B + (size_t)(k + 2 * half) * Nc + n0 + l;
        b.x = bp[0];
        b.y = bp[Nc];
        acc = __builtin_amdgcn_wmma_f32_16x16x4_f32(
                  false, a, false, b, (short)0, acc, false, false);
    }

    float bs = bias ? bias[n0 + l] : 0.f;
#pragma unroll
    for (int r = 0; r < 8; ++r) {
        int row = m0 + r + 8 * half;
        if (row < M) {
            float v = acc[r] + bs;
            if (relu) v = fmaxf(v, 0.f);
            C[(size_t)row * Nc + n0 + l] = v;
        }
    }
}

// ---------------------------------------------------------------------------
// Edge pass 1: per-destination, per-channel max of msg = relu(feat[src]) + eps.
// msg > 0 always, so float-as-int atomicMax is order-correct; table zero-inited.
// One thread per (edge, channel): 16 lanes share an edge -> coalesced feat loads.
// ---------------------------------------------------------------------------
__global__ void gk_edge_max(const long long* __restrict__ ei, long long E,
                            const float* __restrict__ feat,
                            float* __restrict__ segmx)
{
    long long i = (long long)blockIdx.x * blockDim.x + threadIdx.x;
    if (i >= E * 16) return;
    long long e = i >> 4;
    int c = (int)(i & 15);
    int s = (int)ei[e];
    int d = (int)ei[E + e];
    float m = fmaxf(feat[(size_t)s * 16 + c], 0.f) + 1e-7f;
    atomicMax((int*)&segmx[(size_t)d * 16 + c], __float_as_int(m));
}

// Edge pass 2 (fused): denom += exp(m - max), numer += exp(m - max) * m.
__global__ void gk_edge_sum(const long long* __restrict__ ei, long long E,
                            const float* __restrict__ feat,
                            const float* __restrict__ segmx,
                            float* __restrict__ numer,
                            float* __restrict__ denom)
{
    long long i = (long long)blockIdx.x * blockDim.x + threadIdx.x;
    if (i >= E * 16) return;
    long long e = i >> 4;
    int c = (int)(i & 15);
    int s = (int)ei[e];
    int d = (int)ei[E + e];
    float m = fmaxf(feat[(size_t)s * 16 + c], 0.f) + 1e-7f;
    size_t idx = (size_t)d * 16 + c;
    float ex = __expf(m - segmx[idx]);
    atomicAdd(&denom[idx], ex);
    atomicAdd(&numer[idx], ex * m);
}

// aggr = numer/denom (0 for empty segments) + residual feat
__global__ void gk_aggr_add(const float* __restrict__ numer,
                            const float* __restrict__ denom,
                            const float* __restrict__ feat,
                            float* __restrict__ out, int n16)
{
    int i = blockIdx.x * blockDim.x + threadIdx.x;
    if (i >= n16) return;
    float de = denom[i];
    float a = (de > 0.f) ? (numer[i] / de) : 0.f;
    out[i] = a + feat[i];
}

// ---------------------------------------------------------------------------
// BatchNorm stats: deterministic two-pass. Block of 256 = 8 sub-accumulators x
// 32 channels, fixed-order tree -> per-block partial sum / sumsq.
// ---------------------------------------------------------------------------
__global__ void gk_bn_partial(const float* __restrict__ t,
                              float* __restrict__ psum, float* __restrict__ psq,
                              int M, int rowsPerBlock)
{
    __shared__ float ss[256];
    __shared__ float sq[256];
    int c   = threadIdx.x & 31;
    int sub = threadIdx.x >> 5;          // 0..7
    int r0  = blockIdx.x * rowsPerBlock;
    int re  = r0 + rowsPerBlock; if (re > M) re = M;
    float s = 0.f, q = 0.f;
    for (int r = r0 + sub; r < re; r += 8) {
        float v = t[(size_t)r * 32 + c];
        s += v; q += v * v;
    }
    ss[threadIdx.x] = s; sq[threadIdx.x] = q;
    __syncthreads();
    if (sub == 0) {
        for (int k = 1; k < 8; ++k) { s += ss[k * 32 + c]; q += sq[k * 32 + c]; }
        psum[blockIdx.x * 32 + c] = s;
        psq [blockIdx.x * 32 + c] = q;
    }
}

// Finalize: fold gamma/beta into y = h*a[c] + b[c]
__global__ void gk_bn_final(const float* __restrict__ psum, const float* __restrict__ psq,
                            const float* __restrict__ g, const float* __restrict__ be,
                            float* __restrict__ a, float* __restrict__ b,
                            int nblocks, int M)
{
    int c = threadIdx.x;
    if (c >= 32) return;
    float s = 0.f, q = 0.f;
    for (int i = 0; i < nblocks; ++i) { s += psum[i * 32 + c]; q += psq[i * 32 + c]; }
    float mu  = s / (float)M;
    float var = q / (float)M - mu * mu;
    float rs  = rsqrtf(var + 1e-5f);
    float sc  = g[c] * rs;
    a[c] = sc;
    b[c] = be[c] - mu * sc;
}

__global__ void gk_bn_relu(float* __restrict__ t,
                           const float* __restrict__ a, const float* __restrict__ b,
                           int n32)
{
    int i = blockIdx.x * blockDim.x + threadIdx.x;
    if (i >= n32) return;
    int c = i & 31;
    t[i] = fmaxf(t[i] * a[c] + b[c], 0.f);
}

// ---------------------------------------------------------------------------
// Deterministic exclusive scan of node_mask (single block, 1024 threads):
// offs[n] = output slot if mask!=0 else -1.
// ---------------------------------------------------------------------------
__global__ void gk_scan(const int* __restrict__ mask, int* __restrict__ offs, int n)
{
    __shared__ int ssum[1024];
    int tid = threadIdx.x;
    int chunk = (n + 1023) / 1024;
    int s = tid * chunk;
    int e = s + chunk; if (e > n) e = n;
    int acc = 0;
    for (int i = s; i < e; ++i) acc += (mask[i] != 0);
    ssum[tid] = acc;
    __syncthreads();
    for (int off = 1; off < 1024; off <<= 1) {
        int v   = ssum[tid];
        int add = (tid >= off) ? ssum[tid - off] : 0;
        __syncthreads();
        ssum[tid] = v + add;
        __syncthreads();
    }
    int excl = (tid == 0) ? 0 : ssum[tid - 1];
    for (int i = s; i < e; ++i) {
        if (mask[i] != 0) { offs[i] = excl; excl++; }
        else              { offs[i] = -1; }
    }
}

__global__ void gk_gather(const float* __restrict__ fb, const int* __restrict__ offs,
                          float* __restrict__ out, int n32)
{
    int i = blockIdx.x * blockDim.x + threadIdx.x;
    if (i >= n32) return;
    int n = i >> 5, c = i & 31;
    int o = offs[n];
    if (o >= 0) out[(size_t)o * 32 + c] = fb[i];
}

// ---------------------------------------------------------------------------
extern "C" void kernel_launch(void* const* d_in, const int* in_sizes, int n_in,
                              void* d_out, int out_size, void* d_ws, size_t ws_size,
                              hipStream_t stream)
{
    (void)n_in; (void)out_size; (void)ws_size;

    const int N  = in_sizes[4];                       // node_mask length
    const long long E = (long long)in_sizes[1] / 2;   // edge_index (2,E)
    const int INF = in_sizes[3] / N;                  // 128

    const long long* ei = (const long long*)d_in[1];
    const float* x      = (const float*)d_in[3];
    const int*   mask   = (const int*)d_in[4];
    const float* W_src  = (const float*)d_in[5];
    const float* b_src  = (const float*)d_in[6];
    const float* c1_W1  = (const float*)d_in[7];
    const float* c1_b1  = (const float*)d_in[8];
    const float* c1_g   = (const float*)d_in[9];
    const float* c1_be  = (const float*)d_in[10];
    const float* c1_W2  = (const float*)d_in[11];
    const float* c1_b2  = (const float*)d_in[12];
    const float* c2_W1  = (const float*)d_in[13];
    const float* c2_b1  = (const float*)d_in[14];
    const float* c2_g   = (const float*)d_in[15];
    const float* c2_be  = (const float*)d_in[16];
    const float* c2_W2  = (const float*)d_in[17];
    const float* c2_b2  = (const float*)d_in[18];
    const float* fc_W   = (const float*)d_in[19];
    const float* fc_b   = (const float*)d_in[20];
    float* out = (float*)d_out;

    // ---- workspace carve-out ----
    float* ws = (float*)d_ws;
    size_t n16 = (size_t)N * 16, n32 = (size_t)N * 32;
    float* x1    = ws;                 // N x 16 projected features
    float* segmx = x1 + n16;           // N x 16  -- must stay contiguous with
    float* numer = segmx + n16;        // N x 16     numer/denom for one memset
    float* denom = numer + n16;        // N x 16
    float* hb    = denom + n16;        // N x 16 aggr + residual
    float* tb    = hb + n16;           // N x 32 MLP hidden
    float* h1    = tb + n32;           // N x 16 conv output
    float* fb    = h1 + n16;           // N x 32 final pre-mask
    int bnBlocks = (N + 511) / 512;
    float* psum  = fb + n32;
    float* psq   = psum + (size_t)bnBlocks * 32;
    float* bna   = psq + (size_t)bnBlocks * 32;
    float* bnb   = bna + 32;
    int*   offs  = (int*)(bnb + 32);   // N ints

    auto gemm = [&](const float* A, const float* B, const float* bias, float* C,
                    int M, int K, int Nc, int relu) {
        int tiles = ((M + 15) / 16) * (Nc / 16);
        int blocks = (tiles * 32 + 255) / 256;
        gk_gemm_wmma_f32<<<blocks, 256, 0, stream>>>(A, B, bias, C, M, K, Nc, relu);
    };

    long long et = E * 16;
    int eblocks = (int)((et + 255) / 256);
    int nb16 = (int)((n16 + 255) / 256);
    int nb32 = (int)((n32 + 255) / 256);

    auto genconv = [&](const float* feat, const float* W1, const float* b1,
                       const float* g, const float* be,
                       const float* W2, const float* b2,
                       float* hbuf, float* outFeat) {
        hipMemsetAsync(segmx, 0, 3 * n16 * sizeof(float), stream);
        gk_edge_max<<<eblocks, 256, 0, stream>>>(ei, E, feat, segmx);
        gk_edge_sum<<<eblocks, 256, 0, stream>>>(ei, E, feat, segmx, numer, denom);
        gk_aggr_add<<<nb16, 256, 0, stream>>>(numer, denom, feat, hbuf, (int)n16);
        gemm(hbuf, W1, b1, tb, N, 16, 32, 0);
        gk_bn_partial<<<bnBlocks, 256, 0, stream>>>(tb, psum, psq, N, 512);
        gk_bn_final<<<1, 32, 0, stream>>>(psum, psq, g, be, bna, bnb, bnBlocks, N);
        gk_bn_relu<<<nb32, 256, 0, stream>>>(tb, bna, bnb, (int)n32);
        gemm(tb, W2, b2, outFeat, N, 32, 16, 1);  // fuses post-conv relu
    };

    // 1. x1 = x @ W_src + b_src  (K = 128, WMMA main GEMM)
    gemm(x, W_src, b_src, x1, N, INF, 16, 0);
    // 2. conv1 (feat = x1) -> h1
    genconv(x1, c1_W1, c1_b1, c1_g, c1_be, c1_W2, c1_b2, hb, h1);
    // 3. conv2 (feat = h1), reuse x1 buffer for aggr, overwrite h1 with output
    genconv(h1, c2_W1, c2_b1, c2_g, c2_be, c2_W2, c2_b2, x1, h1);
    // 4. final fc: fb = h1 @ fc_W + fc_b
    gemm(h1, fc_W, fc_b, fb, N, 16, 32, 0);
    // 5. masked compaction
    gk_scan<<<1, 1024, 0, stream>>>(mask, offs, N);
    gk_gather<<<nb32, 256, 0, stream>>>(fb, offs, out, (int)n32);
}